// HDAN_2_15470472200900
// MI455X (gfx1250) — compile-verified
//
#include <hip/hip_runtime.h>
#include <hip/hip_bf16.h>

typedef float v2f __attribute__((ext_vector_type(2)));
typedef float v8f __attribute__((ext_vector_type(8)));

__device__ __forceinline__ v8f wmma4(v2f a, v2f b, v8f c) {
  // D = A(16x4,f32) * B(4x16,f32) + C(16x16,f32)  -> v_wmma_f32_16x16x4_f32
  return __builtin_amdgcn_wmma_f32_16x16x4_f32(false, a, false, b, (short)0, c, false, false);
}

// ---------------- prep kernels (weight folding) ----------------

__global__ void k_prep_t1(const float* __restrict__ w, float* __restrict__ wp) {
  for (int i = threadIdx.x; i < 16 * 28; i += blockDim.x) {
    int co = i / 28, k = i % 28;
    wp[i] = (co < 8 && k < 25) ? w[co * 25 + k] : 0.f;
  }
}

// w_e1 [16][8][15]; fold BN1 (scale ss[0..7], shift ss[8..15]) into weights + bias
__global__ void k_prep_e1(const float* __restrict__ w, const float* __restrict__ bias,
                          const float* __restrict__ ss, float* __restrict__ wp,
                          float* __restrict__ cb) {
  for (int i = threadIdx.x; i < 1920; i += blockDim.x) {
    int ci = (i / 15) % 8;
    wp[i] = w[i] * ss[ci];
  }
  if (threadIdx.x < 16) {
    int co = threadIdx.x;
    float s = bias[co];
    for (int ci = 0; ci < 8; ++ci) {
      float t = ss[8 + ci];
      for (int kh = 0; kh < 15; ++kh) s += w[(co * 8 + ci) * 15 + kh] * t;
    }
    cb[co] = s;
  }
}

// w_e2 [16][16][5]; fold psd[kh] (multiplicative only; input already BN'd)
__global__ void k_prep_e2(const float* __restrict__ w, const float* __restrict__ psd,
                          float* __restrict__ wp) {
  for (int i = threadIdx.x; i < 1280; i += blockDim.x) wp[i] = w[i] * psd[i % 5];
}

// ---------------- WMMA conv kernels ----------------

// conv (1x25): X[512,1,55,344] -> Y[512,8,55,320].  A=wpad[16x28], B=patches
__global__ void k_conv_t1(const float* __restrict__ X, const float* __restrict__ wp,
                          const float* __restrict__ bias, float* __restrict__ Y) {
  __shared__ float wl[16 * 28];
  for (int i = threadIdx.x; i < 16 * 28; i += blockDim.x) wl[i] = wp[i];
  __syncthreads();
  int lane = threadIdx.x & 31, wave = threadIdx.x >> 5;
  int m16 = lane & 15, hi = lane >> 4;
  int tile = blockIdx.x * 8 + wave;               // 563200 tiles
  int ttile = tile % 20; int bh = tile / 20;
  int h = bh % 55, b = bh / 55;
  const float* xr = X + (b * 55 + h) * 344;
  int tcol = ttile * 16 + m16;
  v8f acc = {};
  for (int k0 = 0; k0 < 28; k0 += 4) {
    int kx = k0 + 2 * hi;
    v2f a; a.x = wl[m16 * 28 + kx]; a.y = wl[m16 * 28 + kx + 1];
    v2f bb;
    bb.x = (kx < 25) ? xr[tcol + kx] : 0.f;
    bb.y = (kx + 1 < 25) ? xr[tcol + kx + 1] : 0.f;
    acc = wmma4(a, bb, acc);
  }
  if (hi == 0) {
    for (int i = 0; i < 8; ++i)
      Y[((b * 8 + i) * 55 + h) * 320 + tcol] = acc[i] + bias[i];
  }
}

// conv (15x1, stride 10), BN1 folded: Y1[512,8,55,320] -> Y2[512,16,5,320]. K=120
__global__ void k_conv_e1(const float* __restrict__ Y1, const float* __restrict__ wp,
                          const float* __restrict__ cb, float* __restrict__ Y2) {
  __shared__ float wl[16 * 120];
  for (int i = threadIdx.x; i < 16 * 120; i += blockDim.x) wl[i] = wp[i];
  __syncthreads();
  int lane = threadIdx.x & 31, wave = threadIdx.x >> 5;
  int m16 = lane & 15, hi = lane >> 4;
  int tile = blockIdx.x * 8 + wave;               // 51200 tiles
  int ttile = tile % 20; int r = tile / 20;
  int ho = r % 5, b = r / 5;
  int tcol = ttile * 16 + m16;
  const float* yb = Y1 + (size_t)b * 140800;
  v8f acc = {};
  for (int k0 = 0; k0 < 120; k0 += 4) {
    int kx = k0 + 2 * hi;
    v2f a; a.x = wl[m16 * 120 + kx]; a.y = wl[m16 * 120 + kx + 1];
    int ci0 = kx / 15, kh0 = kx - ci0 * 15;
    int k1 = kx + 1, ci1 = k1 / 15, kh1 = k1 - ci1 * 15;
    v2f bb;
    bb.x = yb[(ci0 * 55 + ho * 10 + kh0) * 320 + tcol];
    bb.y = yb[(ci1 * 55 + ho * 10 + kh1) * 320 + tcol];
    acc = wmma4(a, bb, acc);
  }
  for (int i = 0; i < 8; ++i) {
    int co = i + 8 * hi;
    Y2[((b * 16 + co) * 5 + ho) * 320 + tcol] = acc[i] + cb[co];
  }
}

// conv (5x1, stride 5), psd folded: Y2bn[512,16,5,320] -> Y3[512,16,320]. K=80
__global__ void k_conv_e2(const float* __restrict__ Y2, const float* __restrict__ wp,
                          const float* __restrict__ bias, float* __restrict__ Y3) {
  __shared__ float wl[16 * 80];
  for (int i = threadIdx.x; i < 16 * 80; i += blockDim.x) wl[i] = wp[i];
  __syncthreads();
  int lane = threadIdx.x & 31, wave = threadIdx.x >> 5;
  int m16 = lane & 15, hi = lane >> 4;
  int tile = blockIdx.x * 8 + wave;               // 10240 tiles
  int ttile = tile % 20, b = tile / 20;
  int tcol = ttile * 16 + m16;
  const float* yb = Y2 + (size_t)b * 25600;
  v8f acc = {};
  for (int k0 = 0; k0 < 80; k0 += 4) {
    int kx = k0 + 2 * hi;
    v2f a; a.x = wl[m16 * 80 + kx]; a.y = wl[m16 * 80 + kx + 1];
    int ci0 = kx / 5, kh0 = kx - ci0 * 5;
    int k1 = kx + 1, ci1 = k1 / 5, kh1 = k1 - ci1 * 5;
    v2f bb;
    bb.x = yb[(ci0 * 5 + kh0) * 320 + tcol];
    bb.y = yb[(ci1 * 5 + kh1) * 320 + tcol];
    acc = wmma4(a, bb, acc);
  }
  for (int i = 0; i < 8; ++i) {
    int co = i + 8 * hi;
    Y3[(b * 16 + co) * 320 + tcol] = acc[i] + bias[co];
  }
}

// conv (1x10): Y3p[512,16,64] -> Y4[512,16,55]. K=160, guard t
__global__ void k_conv_b1(const float* __restrict__ Y3p, const float* __restrict__ w,
                          const float* __restrict__ bias, float* __restrict__ Y4) {
  __shared__ float wl[16 * 160];
  for (int i = threadIdx.x; i < 16 * 160; i += blockDim.x) wl[i] = w[i];
  __syncthreads();
  int lane = threadIdx.x & 31, wave = threadIdx.x >> 5;
  int m16 = lane & 15, hi = lane >> 4;
  int tile = blockIdx.x * 8 + wave;               // 2048 tiles
  int ttile = tile % 4, b = tile / 4;
  int tcol = ttile * 16 + m16;
  const float* yb = Y3p + (size_t)b * 1024;
  v8f acc = {};
  for (int k0 = 0; k0 < 160; k0 += 4) {
    int kx = k0 + 2 * hi;
    v2f a; a.x = wl[m16 * 160 + kx]; a.y = wl[m16 * 160 + kx + 1];
    int ci0 = kx / 10, dt0 = kx - ci0 * 10;
    int k1 = kx + 1, ci1 = k1 / 10, dt1 = k1 - ci1 * 10;
    int t0 = tcol + dt0, t1 = tcol + dt1;
    v2f bb;
    bb.x = (t0 < 64) ? yb[ci0 * 64 + t0] : 0.f;
    bb.y = (t1 < 64) ? yb[ci1 * 64 + t1] : 0.f;
    acc = wmma4(a, bb, acc);
  }
  if (tcol < 55) {
    for (int i = 0; i < 8; ++i) {
      int co = i + 8 * hi;
      Y4[(b * 16 + co) * 55 + tcol] = acc[i] + bias[co];
    }
  }
}

// conv (2x1, stride 2) merge: yL/yR[512,16,11] -> Y5[512,16,11]. K=32
__global__ void k_conv_e3(const float* __restrict__ YL, const float* __restrict__ YR,
                          const float* __restrict__ w, const float* __restrict__ bias,
                          float* __restrict__ Y5) {
  __shared__ float wl[16 * 32];
  for (int i = threadIdx.x; i < 16 * 32; i += blockDim.x) wl[i] = w[i];
  __syncthreads();
  int lane = threadIdx.x & 31, wave = threadIdx.x >> 5;
  int m16 = lane & 15, hi = lane >> 4;
  int b = blockIdx.x * 8 + wave;                  // 512 tiles
  v8f acc = {};
  for (int k0 = 0; k0 < 32; k0 += 4) {
    int kx = k0 + 2 * hi;
    v2f a; a.x = wl[m16 * 32 + kx]; a.y = wl[m16 * 32 + kx + 1];
    int ci0 = kx >> 1, h0 = kx & 1;
    int k1 = kx + 1, ci1 = k1 >> 1, h1 = k1 & 1;
    const float* s0 = h0 ? YR : YL;
    const float* s1 = h1 ? YR : YL;
    v2f bb;
    bb.x = (m16 < 11) ? s0[(b * 16 + ci0) * 11 + m16] : 0.f;
    bb.y = (m16 < 11) ? s1[(b * 16 + ci1) * 11 + m16] : 0.f;
    acc = wmma4(a, bb, acc);
  }
  if (m16 < 11) {
    for (int i = 0; i < 8; ++i) {
      int co = i + 8 * hi;
      Y5[(b * 16 + co) * 11 + m16] = acc[i] + bias[co];
    }
  }
}

// FC: Out[512,N] = (In[512,K] (*affine)) @ W[N,K]^T + bias
__global__ void k_fc(const float* __restrict__ In, const float* __restrict__ W,
                     const float* __restrict__ bias, const float* __restrict__ ss,
                     float* __restrict__ Out, int K, int N) {
  int lane = threadIdx.x & 31, wave = threadIdx.x >> 5;
  int m16 = lane & 15, hi = lane >> 4;
  int tile = blockIdx.x * 8 + wave;
  int ntiles = N >> 4;
  int nt = tile % ntiles, bt = tile / ntiles;
  int row = bt * 16 + m16;   // batch row for A fragment
  int col = nt * 16 + m16;   // feature col for B fragment
  v8f acc = {};
  for (int k0 = 0; k0 < K; k0 += 4) {
    int kx = k0 + 2 * hi;
    float a0 = In[row * K + kx], a1 = In[row * K + kx + 1];
    if (ss) {
      a0 = a0 * ss[kx] + ss[K + kx];
      a1 = a1 * ss[kx + 1] + ss[K + kx + 1];
    }
    v2f a; a.x = a0; a.y = a1;
    v2f bb; bb.x = W[col * K + kx]; bb.y = W[col * K + kx + 1];
    acc = wmma4(a, bb, acc);
  }
  for (int i = 0; i < 8; ++i) {
    int r = bt * 16 + i + 8 * hi;
    Out[r * N + col] = acc[i] + bias[col];
  }
}

__global__ void k_fc3(const float* __restrict__ in, const float* __restrict__ ss,
                      const float* __restrict__ w, const float* __restrict__ bias,
                      float* __restrict__ out) {
  int i = blockIdx.x * blockDim.x + threadIdx.x;
  if (i >= 1024) return;
  int b = i >> 1, j = i & 1;
  float s = bias[j];
  for (int k = 0; k < 64; ++k)
    s += (in[b * 64 + k] * ss[k] + ss[64 + k]) * w[j * 64 + k];
  out[i] = s;
}

// ---------------- deterministic BN-stat reductions ----------------

__global__ void k_reduce(const float* __restrict__ src, float* __restrict__ part,
                         int C, int S, int chanStride, int outerStride, int nOuter, int P) {
  __shared__ float l1[256], l2[256];
  int c = blockIdx.x / P, p = blockIdx.x % P;
  int total = nOuter * S;
  float sum = 0.f, ssq = 0.f;
  for (int i = p * 256 + threadIdx.x; i < total; i += P * 256) {
    int outer = i / S, inner = i - outer * S;
    float v = src[(size_t)outer * outerStride + (size_t)c * chanStride + inner];
    sum += v; ssq += v * v;
  }
  l1[threadIdx.x] = sum; l2[threadIdx.x] = ssq;
  __syncthreads();
  for (int w = 128; w > 0; w >>= 1) {
    if (threadIdx.x < w) { l1[threadIdx.x] += l1[threadIdx.x + w]; l2[threadIdx.x] += l2[threadIdx.x + w]; }
    __syncthreads();
  }
  if (threadIdx.x == 0) { part[(c * P + p) * 2] = l1[0]; part[(c * P + p) * 2 + 1] = l2[0]; }
}

__global__ void k_finalize_bn(const float* __restrict__ part, float* __restrict__ ssOut,
                              const float* __restrict__ g, const float* __restrict__ be,
                              int C, int P, float invN, float eps) {
  int c = threadIdx.x;
  if (c < C) {
    float sum = 0.f, ssq = 0.f;
    for (int p = 0; p < P; ++p) { sum += part[(c * P + p) * 2]; ssq += part[(c * P + p) * 2 + 1]; }
    float mean = sum * invN;
    float var = ssq * invN - mean * mean;
    float sc = g[c] * rsqrtf(var + eps);
    ssOut[c] = sc;
    ssOut[C + c] = be[c] - mean * sc;
  }
}

// psd[h] = 320 * sum_{b,c,t} x^2  (Parseval: replaces the FFT)
__global__ void k_finalize_psd(const float* __restrict__ part, float* __restrict__ psd, int P) {
  int h = threadIdx.x;
  if (h < 5) {
    float ssq = 0.f;
    for (int p = 0; p < P; ++p) ssq += part[(h * P + p) * 2 + 1];
    psd[h] = 320.f * ssq;
  }
}

// ---------------- elementwise ----------------

__global__ void k_bnapply_y2(float* __restrict__ y, const float* __restrict__ ss) {
  const int total = 8192000;
  for (int i = blockIdx.x * blockDim.x + threadIdx.x; i < total; i += gridDim.x * blockDim.x) {
    int c = (i / 1600) & 15;
    y[i] = y[i] * ss[c] + ss[16 + c];
  }
}

// dst[(b*C+c)*Tout+u] = mean_{j<pk} elu(src[(b*C+c)*Tin + u*pk+j]*s[c]+t[c])
__global__ void k_bn_elu_pool(const float* __restrict__ src, float* __restrict__ dst,
                              const float* __restrict__ ss, int C, int Tin, int Tout,
                              int pk, int n) {
  int i = blockIdx.x * blockDim.x + threadIdx.x;
  if (i >= n) return;
  int u = i % Tout, bc = i / Tout, c = bc % C;
  float sc = ss[c], sh = ss[C + c];
  float acc = 0.f;
  for (int j = 0; j < pk; ++j) {
    float v = src[bc * Tin + u * pk + j] * sc + sh;
    acc += (v > 0.f) ? v : expm1f(v);
  }
  dst[i] = acc / (float)pk;
}

// ---------------- host ----------------

extern "C" void kernel_launch(void* const* d_in, const int* in_sizes, int n_in,
                              void* d_out, int out_size, void* d_ws, size_t ws_size,
                              hipStream_t stream) {
  (void)in_sizes; (void)n_in; (void)out_size; (void)ws_size;
  const float* srcL   = (const float*)d_in[0];
  const float* srcR   = (const float*)d_in[1];
  const float* w_t1   = (const float*)d_in[2];
  const float* b_t1   = (const float*)d_in[3];
  const float* g_t1   = (const float*)d_in[4];
  const float* be_t1  = (const float*)d_in[5];
  const float* w_e1   = (const float*)d_in[6];
  const float* b_e1   = (const float*)d_in[7];
  const float* g_e1   = (const float*)d_in[8];
  const float* be_e1  = (const float*)d_in[9];
  const float* w_e2   = (const float*)d_in[10];
  const float* b_e2   = (const float*)d_in[11];
  const float* g_e2   = (const float*)d_in[12];
  const float* be_e2  = (const float*)d_in[13];
  const float* w_b1   = (const float*)d_in[14];
  const float* b_b1   = (const float*)d_in[15];
  const float* g_b1   = (const float*)d_in[16];
  const float* be_b1  = (const float*)d_in[17];
  const float* w_e3   = (const float*)d_in[18];
  const float* b_e3   = (const float*)d_in[19];
  const float* g_e3   = (const float*)d_in[20];
  const float* be_e3  = (const float*)d_in[21];
  const float* w_fc1  = (const float*)d_in[22];
  const float* b_fc1  = (const float*)d_in[23];
  const float* g_fc1  = (const float*)d_in[24];
  const float* be_fc1 = (const float*)d_in[25];
  const float* w_f1   = (const float*)d_in[26];
  const float* b_f1   = (const float*)d_in[27];
  const float* g_f1   = (const float*)d_in[28];
  const float* be_f1  = (const float*)d_in[29];
  const float* w_f2   = (const float*)d_in[30];
  const float* b_f2   = (const float*)d_in[31];

  float* ws = (float*)d_ws;
  // workspace layout (floats)
  constexpr size_t OFF_Y1  = 0;                       // 72,089,600
  constexpr size_t OFF_Y2  = OFF_Y1 + 72089600;       //  8,192,000
  constexpr size_t OFF_Y3  = OFF_Y2 + 8192000;        //  2,621,440
  constexpr size_t OFF_Y3P = OFF_Y3 + 2621440;        //    524,288
  constexpr size_t OFF_Y4  = OFF_Y3P + 524288;        //    450,560
  constexpr size_t OFF_YLP = OFF_Y4 + 450560;         //     90,112
  constexpr size_t OFF_YRP = OFF_YLP + 90112;         //     90,112
  constexpr size_t OFF_Y5  = OFF_YRP + 90112;         //     90,112
  constexpr size_t OFF_Z   = OFF_Y5 + 90112;          //     90,112
  constexpr size_t OFF_FC1 = OFF_Z + 90112;           //    131,072
  constexpr size_t OFF_FC2 = OFF_FC1 + 131072;        //     32,768
  constexpr size_t OFF_PRT = OFF_FC2 + 32768;         //     32,768
  constexpr size_t SS1     = OFF_PRT + 32768;         // 16
  constexpr size_t SS2     = SS1 + 16;                // 32
  constexpr size_t SS3     = SS2 + 32;                // 32
  constexpr size_t SS4     = SS3 + 32;                // 32
  constexpr size_t SS5     = SS4 + 32;                // 32
  constexpr size_t SS6     = SS5 + 32;                // 512
  constexpr size_t SS7     = SS6 + 512;               // 128
  constexpr size_t PSD     = SS7 + 128;               // 8
  constexpr size_t WP_T1   = PSD + 8;                 // 448
  constexpr size_t WP_E1   = WP_T1 + 448;             // 1920
  constexpr size_t CB_E1   = WP_E1 + 1920;            // 16
  constexpr size_t WP_E2   = CB_E1 + 16;              // 1280

  float* part = ws + OFF_PRT;

  k_prep_t1<<<1, 256, 0, stream>>>(w_t1, ws + WP_T1);

  for (int br = 0; br < 2; ++br) {
    const float* X = br ? srcR : srcL;
    float* ypf = ws + (br ? OFF_YRP : OFF_YLP);

    // conv_t1 + BN1 stats
    k_conv_t1<<<70400, 256, 0, stream>>>(X, ws + WP_T1, b_t1, ws + OFF_Y1);
    k_reduce<<<8 * 64, 256, 0, stream>>>(ws + OFF_Y1, part, 8, 17600, 17600, 140800, 512, 64);
    k_finalize_bn<<<1, 256, 0, stream>>>(part, ws + SS1, g_t1, be_t1, 8, 64, 1.f / 9011200.f, 0.f);

    // conv_e1 (BN1 folded) + BN2 stats + apply + psd
    k_prep_e1<<<1, 256, 0, stream>>>(w_e1, b_e1, ws + SS1, ws + WP_E1, ws + CB_E1);
    k_conv_e1<<<6400, 256, 0, stream>>>(ws + OFF_Y1, ws + WP_E1, ws + CB_E1, ws + OFF_Y2);
    k_reduce<<<16 * 64, 256, 0, stream>>>(ws + OFF_Y2, part, 16, 1600, 1600, 25600, 512, 64);
    k_finalize_bn<<<1, 256, 0, stream>>>(part, ws + SS2, g_e1, be_e1, 16, 64, 1.f / 819200.f, 0.f);
    k_bnapply_y2<<<8192, 256, 0, stream>>>(ws + OFF_Y2, ws + SS2);
    k_reduce<<<5 * 64, 256, 0, stream>>>(ws + OFF_Y2, part, 5, 320, 320, 1600, 8192, 64);
    k_finalize_psd<<<1, 32, 0, stream>>>(part, ws + PSD, 64);

    // conv_e2 (psd folded) + BN3 + ELU + pool5
    k_prep_e2<<<1, 256, 0, stream>>>(w_e2, ws + PSD, ws + WP_E2);
    k_conv_e2<<<1280, 256, 0, stream>>>(ws + OFF_Y2, ws + WP_E2, b_e2, ws + OFF_Y3);
    k_reduce<<<16 * 64, 256, 0, stream>>>(ws + OFF_Y3, part, 16, 320, 320, 5120, 512, 64);
    k_finalize_bn<<<1, 256, 0, stream>>>(part, ws + SS3, g_e2, be_e2, 16, 64, 1.f / 163840.f, 0.f);
    k_bn_elu_pool<<<2048, 256, 0, stream>>>(ws + OFF_Y3, ws + OFF_Y3P, ws + SS3, 16, 320, 64, 5, 524288);

    // conv_b1 + BN4 + ELU + pool5
    k_conv_b1<<<256, 256, 0, stream>>>(ws + OFF_Y3P, w_b1, b_b1, ws + OFF_Y4);
    k_reduce<<<16 * 64, 256, 0, stream>>>(ws + OFF_Y4, part, 16, 55, 55, 880, 512, 64);
    k_finalize_bn<<<1, 256, 0, stream>>>(part, ws + SS4, g_b1, be_b1, 16, 64, 1.f / 28160.f, 0.f);
    k_bn_elu_pool<<<352, 256, 0, stream>>>(ws + OFF_Y4, ypf, ws + SS4, 16, 55, 11, 5, 90112);
  }

  // merge: conv_e3 + BN5 + ELU -> z[512,176]
  k_conv_e3<<<64, 256, 0, stream>>>(ws + OFF_YLP, ws + OFF_YRP, w_e3, b_e3, ws + OFF_Y5);
  k_reduce<<<16 * 64, 256, 0, stream>>>(ws + OFF_Y5, part, 16, 11, 11, 176, 512, 64);
  k_finalize_bn<<<1, 256, 0, stream>>>(part, ws + SS5, g_e3, be_e3, 16, 64, 1.f / 5632.f, 0.f);
  k_bn_elu_pool<<<352, 256, 0, stream>>>(ws + OFF_Y5, ws + OFF_Z, ws + SS5, 16, 11, 11, 1, 90112);

  // FC1 (176->256) + BN1d
  k_fc<<<64, 256, 0, stream>>>(ws + OFF_Z, w_fc1, b_fc1, nullptr, ws + OFF_FC1, 176, 256);
  k_reduce<<<256 * 2, 256, 0, stream>>>(ws + OFF_FC1, part, 256, 1, 1, 256, 512, 2);
  k_finalize_bn<<<1, 256, 0, stream>>>(part, ws + SS6, g_fc1, be_fc1, 256, 2, 1.f / 512.f, 1e-5f);

  // FC2 (256->64, affine at load) + BN1d
  k_fc<<<16, 256, 0, stream>>>(ws + OFF_FC1, w_f1, b_f1, ws + SS6, ws + OFF_FC2, 256, 64);
  k_reduce<<<64 * 2, 256, 0, stream>>>(ws + OFF_FC2, part, 64, 1, 1, 64, 512, 2);
  k_finalize_bn<<<1, 256, 0, stream>>>(part, ws + SS7, g_f1, be_f1, 64, 2, 1.f / 512.f, 1e-5f);

  // FC3 (64->2)
  k_fc3<<<4, 256, 0, stream>>>(ws + OFF_FC2, ws + SS7, w_f2, b_f2, (float*)d_out);
}